// GritLMPooler_11192684773505
// MI455X (gfx1250) — compile-verified
//
#include <hip/hip_runtime.h>
#include <hip/hip_bf16.h>
#include <math.h>

typedef float v2f __attribute__((ext_vector_type(2)));
typedef float v8f __attribute__((ext_vector_type(8)));

#define UNROLL 8

__device__ __forceinline__ v8f wmma_step(int ta, int lo, int hi,
                                         float bx, float by, v8f c) {
    v2f a, b;
    b.x = bx; b.y = by;
    a.x = (ta     >= lo && ta     < hi) ? 1.0f : 0.0f;
    a.y = (ta + 1 >= lo && ta + 1 < hi) ? 1.0f : 0.0f;
    return __builtin_amdgcn_wmma_f32_16x16x4_f32(
        false, a, false, b, (short)0, c, false, false);
}

// ---------------------------------------------------------------------------
// Kernel 1: per (column-tile, K-chunk) wave computes a 16x16 partial of
//   sums[seg, dim] = sum_t mask(t,seg) * hs[t, dim]
// via V_WMMA_F32_16X16X4_F32. A = 16x4 0/1 mask tile (computed in regs),
// B = 4x16 hidden tile (NT-loaded). Unrolled x8: 16 loads in flight, then
// 8 WMMAs alternating between two accumulator chains.
// ---------------------------------------------------------------------------
__global__ __launch_bounds__(32)
void gritlm_pool_wmma_partial(const float* __restrict__ hs,
                              const int* __restrict__ plens,
                              const int* __restrict__ ilens,
                              float* __restrict__ partial,
                              int B, int D, int TOTAL,
                              int chunk_tokens, int nTiles) {
    const int tile  = blockIdx.x;
    const int chunk = blockIdx.y;
    const int lane  = threadIdx.x;     // 0..31
    const int n     = lane & 15;       // dim within tile / segment id (A row)
    const int half  = lane >> 4;       // 0: K=0,1   1: K=2,3

    // Per-lane segment window: token t hits segment n iff lo <= t < hi.
    int lo, hi;
    if (n < B) {
        int e = 0;
        for (int i = 0; i <= n; ++i) e += plens[i];
        lo = e - plens[n] + ilens[n];
        hi = e;
    } else {
        lo = 1; hi = 0;                // always-zero mask row
    }

    const int t_begin = chunk * chunk_tokens;
    const int t_end   = t_begin + chunk_tokens;
    const int Tfloor  = TOTAL & ~3;
    const int t_main  = (t_end < Tfloor) ? t_end : Tfloor;   // unguarded region

    // Lane base pointer: token (t_begin + 2*half), dim (tile*16 + n).
    const float* p = hs + (size_t)(t_begin + 2 * half) * D + (size_t)tile * 16 + n;
    const size_t rowStep = (size_t)D;          // elements per token row

    v8f c0 = {0.f,0.f,0.f,0.f,0.f,0.f,0.f,0.f};
    v8f c1 = {0.f,0.f,0.f,0.f,0.f,0.f,0.f,0.f};

    int t0 = t_begin;

    // ---- main unrolled loop: no guards, 16 loads then 8 WMMAs ----
    for (; t0 + 4 * UNROLL <= t_main; t0 += 4 * UNROLL) {
        float bx[UNROLL], by[UNROLL];
#pragma unroll
        for (int u = 0; u < UNROLL; ++u) {
            bx[u] = __builtin_nontemporal_load(p + (size_t)(4 * u) * rowStep);
            by[u] = __builtin_nontemporal_load(p + (size_t)(4 * u + 1) * rowStep);
        }
#pragma unroll
        for (int u = 0; u < UNROLL; ++u) {
            const int ta = t0 + 4 * u + 2 * half;
            if (u & 1) c1 = wmma_step(ta, lo, hi, bx[u], by[u], c1);
            else       c0 = wmma_step(ta, lo, hi, bx[u], by[u], c0);
        }
        p += (size_t)(4 * UNROLL) * rowStep;
    }

    // ---- leftover full 4-token groups (still unguarded) ----
    for (; t0 + 4 <= t_main; t0 += 4) {
        float bx = __builtin_nontemporal_load(p);
        float by = __builtin_nontemporal_load(p + rowStep);
        c0 = wmma_step(t0 + 2 * half, lo, hi, bx, by, c0);
        p += 4 * rowStep;
    }

    // ---- tail (<=1 group, only in the chunk containing end-of-data) ----
    if (t0 < t_end && t0 < TOTAL) {
        const int ta = t0 + 2 * half;
        float bx = 0.f, by = 0.f;
        if (ta < TOTAL)     bx = __builtin_nontemporal_load(p);
        if (ta + 1 < TOTAL) by = __builtin_nontemporal_load(p + rowStep);
        c0 = wmma_step(ta, lo, hi, bx, by, c0);
    }

    const v8f c = c0 + c1;

    // Store 16x16 partial: C VGPR r holds (M = r + 8*half, N = n).
    const size_t base = ((size_t)chunk * nTiles + tile) * 256;
    const int mrow0 = 8 * half;
#pragma unroll
    for (int r = 0; r < 8; ++r) {
        partial[base + (size_t)(mrow0 + r) * 16 + n] = c[r];
    }
}

// ---------------------------------------------------------------------------
// Kernel 2: one block per sequence. Sum partials over chunks, divide by
// count, L2-normalize, write out. Deterministic tree reduction.
// ---------------------------------------------------------------------------
__global__ __launch_bounds__(256)
void gritlm_pool_finalize(const float* __restrict__ partial,
                          const int* __restrict__ plens,
                          const int* __restrict__ ilens,
                          float* __restrict__ out,
                          int D, int nChunks, int nTiles) {
    const int m   = blockIdx.x;
    const int tid = threadIdx.x;

    const float invCnt = 1.0f / (float)(plens[m] - ilens[m]);

    __shared__ float red[256];

    float sumsq = 0.0f;
    for (int d = tid; d < D; d += 256) {
        const int tile = d >> 4;
        const int nn   = d & 15;
        float s = 0.0f;
        for (int ch = 0; ch < nChunks; ++ch) {
            s += partial[((size_t)ch * nTiles + tile) * 256 + (size_t)m * 16 + nn];
        }
        const float mean = s * invCnt;
        out[(size_t)m * D + d] = mean;
        sumsq += mean * mean;
    }

    red[tid] = sumsq;
    __syncthreads();
#pragma unroll
    for (int off = 128; off > 0; off >>= 1) {
        if (tid < off) red[tid] += red[tid + off];
        __syncthreads();
    }
    const float invNorm = 1.0f / fmaxf(sqrtf(red[0]), 1e-12f);
    __syncthreads();

    for (int d = tid; d < D; d += 256) {
        out[(size_t)m * D + d] *= invNorm;
    }
}

extern "C" void kernel_launch(void* const* d_in, const int* in_sizes, int n_in,
                              void* d_out, int out_size, void* d_ws, size_t ws_size,
                              hipStream_t stream) {
    const float* hs    = (const float*)d_in[0];
    const int*   plens = (const int*)d_in[1];
    const int*   ilens = (const int*)d_in[2];
    float*       out   = (float*)d_out;

    const int B = in_sizes[1];                     // 16
    const int D = out_size / B;                    // 4096
    const int TOTAL = in_sizes[0] / D;             // 32768
    const int nTiles = D / 16;                     // 256

    // Workspace: nChunks * nTiles * 256 floats. Target 64 chunks, shrink to ws.
    const size_t bytesPerChunk = (size_t)nTiles * 256 * sizeof(float);
    int nChunks = 64;
    const size_t maxChunks = ws_size / bytesPerChunk;
    if ((size_t)nChunks > maxChunks) nChunks = (int)maxChunks;
    if (nChunks < 1) nChunks = 1;

    int chunk_tokens = (TOTAL + nChunks - 1) / nChunks;
    chunk_tokens = (chunk_tokens + 3) & ~3;                      // multiple of 4
    nChunks = (TOTAL + chunk_tokens - 1) / chunk_tokens;         // drop empties

    float* partial = (float*)d_ws;

    dim3 grid1(nTiles, nChunks);
    gritlm_pool_wmma_partial<<<grid1, 32, 0, stream>>>(
        hs, plens, ilens, partial, B, D, TOTAL, chunk_tokens, nTiles);

    gritlm_pool_finalize<<<B, 256, 0, stream>>>(
        partial, plens, ilens, out, D, nChunks, nTiles);
}